// SegmentationBranch_41583873360672
// MI455X (gfx1250) — compile-verified
//
#include <hip/hip_runtime.h>
#include <hip/hip_bf16.h>
#include <math.h>

typedef __attribute__((ext_vector_type(2))) float v2f;
typedef __attribute__((ext_vector_type(8))) float v8f;

#define HID  64
#define NCLS 11      // NUM_CLASSES + 1
#define KIN  512
#define OH   480
#define OW   640
#define LH   60
#define LW   80
#define BATCH 4

// ---------------------------------------------------------------------------
// conv1x1 512->64 + bias + ReLU via V_WMMA_F32_16X16X4_F32.
// X: (B,512,HW) NCHW, W: (64,512), Y: (B,64,HW).
// grid.x = B * (HW/16); block = 128 = 4 waves; wave w computes out-ch tile w*16.
// A layout (ISA 7.12.2, 32-bit A 16x4): lane L holds M=L&15, K = v + 2*(L>>4).
// B layout mirrored: lane L holds N=L&15, K = v + 2*(L>>4).
// D layout: VGPR v, lane L -> M = v + 8*(L>>4), N = L&15.
// ---------------------------------------------------------------------------
__global__ void conv512to64_wmma(const float* __restrict__ X,
                                 const float* __restrict__ W,
                                 const float* __restrict__ bias,
                                 float* __restrict__ Y,
                                 int HW, int ntiles)
{
    int wave = threadIdx.x >> 5;
    int lane = threadIdx.x & 31;
    int ln = lane & 15;
    int lh = lane >> 4;                    // 0 or 1
    int b  = blockIdx.x / ntiles;
    int pt = (blockIdx.x % ntiles) * 16;
    int m0 = wave * 16;

    const float* Wr = W + (size_t)(m0 + ln) * KIN + 2 * lh;       // A row
    const float* Xc = X + (size_t)b * KIN * HW + (size_t)(2 * lh) * HW + pt + ln;

    v8f acc = {};
    for (int k = 0; k < KIN; k += 4) {
        v2f a = *(const v2f*)(Wr + k);                            // K=k+2lh, k+1+2lh
        v2f bv;
        bv.x = Xc[(size_t)k * HW];
        bv.y = Xc[(size_t)(k + 1) * HW];
        acc = __builtin_amdgcn_wmma_f32_16x16x4_f32(false, a, false, bv,
                                                    (short)0, acc, false, false);
    }
#pragma unroll
    for (int v = 0; v < 8; ++v) {
        int oc = m0 + v + 8 * lh;
        float r = acc[v] + bias[oc];
        r = r > 0.f ? r : 0.f;
        Y[((size_t)b * HID + oc) * HW + pt + ln] = r;
    }
}

// ---------------------------------------------------------------------------
// feat_low = f1 + bilinear_x2(f2)  (half-pixel, edge clamp), stored NHWC
// so the classifier can do contiguous per-pixel channel loads.
// ---------------------------------------------------------------------------
__global__ void fuse_upsample2_add(const float* __restrict__ f1,   // (B,64,4800)
                                   const float* __restrict__ f2,   // (B,64,1200)
                                   float* __restrict__ feat,       // (B,4800,64)
                                   int total)
{
    int tid = blockIdx.x * blockDim.x + threadIdx.x;
    if (tid >= total) return;
    int c   = tid & 63;
    int pix = (tid >> 6) % (LH * LW);
    int b   = tid / ((LH * LW) * HID);
    int y = pix / LW, x = pix % LW;
    float sy = y * 0.5f - 0.25f;                       // (y+0.5)/2 - 0.5
    float sx = x * 0.5f - 0.25f;
    int iy0 = (int)floorf(sy), ix0 = (int)floorf(sx);
    float fy = sy - (float)iy0, fx = sx - (float)ix0;
    int iy1 = min(max(iy0 + 1, 0), LH / 2 - 1);
    int ix1 = min(max(ix0 + 1, 0), LW / 2 - 1);
    iy0 = min(max(iy0, 0), LH / 2 - 1);
    ix0 = min(max(ix0, 0), LW / 2 - 1);
    const float* p = f2 + ((size_t)b * HID + c) * (LH / 2) * (LW / 2);
    float v00 = p[iy0 * (LW / 2) + ix0], v01 = p[iy0 * (LW / 2) + ix1];
    float v10 = p[iy1 * (LW / 2) + ix0], v11 = p[iy1 * (LW / 2) + ix1];
    float up = v00 * (1.f - fy) * (1.f - fx) + v01 * (1.f - fy) * fx
             + v10 * fy * (1.f - fx)         + v11 * fy * fx;
    feat[tid] = f1[((size_t)b * HID + c) * (LH * LW) + pix] + up;
}

// ---------------------------------------------------------------------------
// Fused x8 bilinear upsample + 11x64 classifier (WMMA) + softmax + argmax.
// One wave per 16-pixel tile of one output row. A = wc padded to 16x64,
// B = interpolated 64x16 feature columns (built in registers, never hits HBM).
// Logits for pixel n live in lanes n (classes 0-7) and n+16 (classes 8-15);
// softmax reduction is a single shfl_xor(16) pair exchange.
// ---------------------------------------------------------------------------
__global__ void classify_softmax_wmma(const float* __restrict__ feat, // (B,4800,64)
                                      const float* __restrict__ wc,   // (11,64)
                                      const float* __restrict__ bc,   // (11)
                                      float* __restrict__ prob,       // (B,11,480,640)
                                      float* __restrict__ segf,       // (B,480,640)
                                      int*   __restrict__ segi)
{
    int wave = threadIdx.x >> 5;
    int lane = threadIdx.x & 31;
    int ln = lane & 15, lh = lane >> 4;
    int t = blockIdx.x * 4 + wave;
    int b = t / (OH * (OW / 16));
    int r = t % (OH * (OW / 16));
    int y = r / (OW / 16);
    int x = (r % (OW / 16)) * 16 + ln;

    float sy = (y + 0.5f) * 0.125f - 0.5f;
    float sx = (x + 0.5f) * 0.125f - 0.5f;
    int iy0 = (int)floorf(sy), ix0 = (int)floorf(sx);
    float fy = sy - (float)iy0, fx = sx - (float)ix0;
    int iy1 = min(max(iy0 + 1, 0), LH - 1);
    int ix1 = min(max(ix0 + 1, 0), LW - 1);
    iy0 = min(max(iy0, 0), LH - 1);
    ix0 = min(max(ix0, 0), LW - 1);
    float w00 = (1.f - fy) * (1.f - fx), w01 = (1.f - fy) * fx;
    float w10 = fy * (1.f - fx),         w11 = fy * fx;

    const float* base = feat + (size_t)b * (LH * LW) * HID;
    const float* p00 = base + (iy0 * LW + ix0) * HID;
    const float* p01 = base + (iy0 * LW + ix1) * HID;
    const float* p10 = base + (iy1 * LW + ix0) * HID;
    const float* p11 = base + (iy1 * LW + ix1) * HID;

    int cls = ln;                            // A-matrix row this lane feeds
    v8f acc = {};
    for (int k = 0; k < HID; k += 4) {
        int c0 = k + 2 * lh;
        v2f a;
        if (cls < NCLS) a = *(const v2f*)(wc + cls * HID + c0);
        else { a.x = 0.f; a.y = 0.f; }
        v2f f00 = *(const v2f*)(p00 + c0);
        v2f f01 = *(const v2f*)(p01 + c0);
        v2f f10 = *(const v2f*)(p10 + c0);
        v2f f11 = *(const v2f*)(p11 + c0);
        v2f bv;
        bv.x = f00.x * w00 + f01.x * w01 + f10.x * w10 + f11.x * w11;
        bv.y = f00.y * w00 + f01.y * w01 + f10.y * w10 + f11.y * w11;
        acc = __builtin_amdgcn_wmma_f32_16x16x4_f32(false, a, false, bv,
                                                    (short)0, acc, false, false);
    }

    // logits for class cv = v + 8*lh at pixel x
    float lg[8], ex[8];
    float mx = -3.0e38f;
#pragma unroll
    for (int v = 0; v < 8; ++v) {
        int cv = v + 8 * lh;
        float l = (cv < NCLS) ? (acc[v] + bc[cv]) : -3.0e38f;
        lg[v] = l;
        mx = fmaxf(mx, l);
    }
    mx = fmaxf(mx, __shfl_xor(mx, 16));
    float s = 0.f;
#pragma unroll
    for (int v = 0; v < 8; ++v) {
        int cv = v + 8 * lh;
        float e = (cv < NCLS) ? __expf(lg[v] - mx) : 0.f;
        ex[v] = e; s += e;
    }
    s += __shfl_xor(s, 16);
    float inv = 1.f / s;

    size_t pixoff = ((size_t)b * OH + y) * OW + x;
#pragma unroll
    for (int v = 0; v < 8; ++v) {
        int cv = v + 8 * lh;
        if (cv < NCLS)
            prob[((size_t)b * NCLS + cv) * (OH * OW) + (size_t)y * OW + x] = ex[v] * inv;
    }
    // argmax, first-max-wins (matches jnp.argmax)
    float bv2 = -3.0e38f; int bi = 0;
#pragma unroll
    for (int v = 0; v < 8; ++v) {
        int cv = v + 8 * lh;
        if (cv < NCLS && lg[v] > bv2) { bv2 = lg[v]; bi = cv; }
    }
    float ov = __shfl_xor(bv2, 16);
    int   oi = __shfl_xor(bi, 16);
    if (ov > bv2 || (ov == bv2 && oi < bi)) { bv2 = ov; bi = oi; }
    if (lh == 0) {
        segf[pixoff] = (float)bi;
        segi[pixoff] = bi;
    }
}

// ---------------------------------------------------------------------------
// bbox: per-(b, class 1..9) count / min / max via LDS reduction + global merge
// ---------------------------------------------------------------------------
__global__ void bbox_init(int* __restrict__ g)      // 36 entries x 5
{
    int i = threadIdx.x;
    if (i < BATCH * 9) {
        g[i * 5 + 0] = 0;       // count
        g[i * 5 + 1] = OW;      // x1 (min)
        g[i * 5 + 2] = OH;      // y1 (min)
        g[i * 5 + 3] = -1;      // x2 (max)
        g[i * 5 + 4] = -1;      // y2 (max)
    }
}

__global__ void bbox_accum(const int* __restrict__ seg, int* __restrict__ g)
{
    __shared__ int s[9 * 5];
    int tid = threadIdx.x;
    if (tid < 45) {
        int f = tid % 5;
        s[tid] = (f == 0) ? 0 : (f == 1 ? OW : (f == 2 ? OH : -1));
    }
    __syncthreads();
    int pid = blockIdx.x * blockDim.x + tid;     // 307200 % 256 == 0 -> one b/block
    int b = pid / (OH * OW);
    int rem = pid % (OH * OW);
    int y = rem / OW, x = rem % OW;
    int cls = seg[pid];
    if (cls >= 1 && cls <= 9) {
        int k = cls - 1;
        atomicAdd(&s[k * 5 + 0], 1);
        atomicMin(&s[k * 5 + 1], x);
        atomicMin(&s[k * 5 + 2], y);
        atomicMax(&s[k * 5 + 3], x);
        atomicMax(&s[k * 5 + 4], y);
    }
    __syncthreads();
    if (tid < 45) {
        int k = tid / 5, f = tid % 5;
        int v = s[tid];
        int* gp = &g[(b * 9 + k) * 5 + f];
        if (f == 0)      { if (v) atomicAdd(gp, v); }
        else if (f <= 2) atomicMin(gp, v);
        else             atomicMax(gp, v);
    }
}

__global__ void bbox_final(const int* __restrict__ g, float* __restrict__ out)
{
    int i = threadIdx.x;
    if (i >= BATCH * 9) return;
    int b = i / 9, k = i % 9;
    const int* e = &g[i * 5];
    float* row = out + i * 6;
    if (e[0] >= 500) {
        row[0] = (float)b;
        row[1] = (float)e[1];
        row[2] = (float)e[2];
        row[3] = (float)e[3];
        row[4] = (float)e[4];
        row[5] = (float)(k + 1);
    } else {
        for (int j = 0; j < 6; ++j) row[j] = -1.f;
    }
}

// ---------------------------------------------------------------------------
extern "C" void kernel_launch(void* const* d_in, const int* in_sizes, int n_in,
                              void* d_out, int out_size, void* d_ws, size_t ws_size,
                              hipStream_t stream)
{
    const float* feature1 = (const float*)d_in[0];  // (4,512,60,80)
    const float* feature2 = (const float*)d_in[1];  // (4,512,30,40)
    const float* w1 = (const float*)d_in[2];
    const float* b1 = (const float*)d_in[3];
    const float* w2 = (const float*)d_in[4];
    const float* b2 = (const float*)d_in[5];
    const float* wc = (const float*)d_in[6];
    const float* bc = (const float*)d_in[7];

    float* prob = (float*)d_out;                                   // 4*11*480*640
    float* segf = prob + (size_t)BATCH * NCLS * OH * OW;           // 4*480*640
    float* bbx  = segf + (size_t)BATCH * OH * OW;                  // 36*6

    char* ws = (char*)d_ws;
    float* f1   = (float*)ws;  ws += (size_t)BATCH * HID * (LH * LW) * 4;
    float* f2   = (float*)ws;  ws += (size_t)BATCH * HID * (LH * LW / 4) * 4;
    float* feat = (float*)ws;  ws += (size_t)BATCH * (LH * LW) * HID * 4;
    int*   segi = (int*)ws;    ws += (size_t)BATCH * OH * OW * 4;
    int*   gacc = (int*)ws;    // 36*5 ints

    conv512to64_wmma<<<BATCH * (LH * LW / 16), 128, 0, stream>>>(
        feature1, w1, b1, f1, LH * LW, LH * LW / 16);
    conv512to64_wmma<<<BATCH * (LH * LW / 64), 128, 0, stream>>>(
        feature2, w2, b2, f2, LH * LW / 4, LH * LW / 64);

    int totalC = BATCH * (LH * LW) * HID;
    fuse_upsample2_add<<<(totalC + 255) / 256, 256, 0, stream>>>(f1, f2, feat, totalC);

    classify_softmax_wmma<<<BATCH * OH * (OW / 16) / 4, 128, 0, stream>>>(
        feat, wc, bc, prob, segf, segi);

    bbox_init<<<1, 64, 0, stream>>>(gacc);
    bbox_accum<<<BATCH * OH * OW / 256, 256, 0, stream>>>(segi, gacc);
    bbox_final<<<1, 64, 0, stream>>>(gacc, bbx);
}